// MultiHeadDampedEMA_19112604467662
// MI455X (gfx1250) — compile-verified
//
#include <hip/hip_runtime.h>

// Problem constants (from reference): L=4096, B=8, D=1024, N=16
#define LL 4096
#define BB 8
#define DD 1024
#define NN 16
#define CCH 16            // chunk size == WMMA tile dim
#define NC (LL / CCH)     // 256 chunks

typedef float v2f __attribute__((ext_vector_type(2)));
typedef float v8f __attribute__((ext_vector_type(8)));
typedef unsigned int v4u __attribute__((ext_vector_type(4)));
typedef int v4i __attribute__((ext_vector_type(4)));
typedef int v8i __attribute__((ext_vector_type(8)));

__device__ __forceinline__ float sigmoidf_(float v) {
    return 1.0f / (1.0f + __expf(-v));
}

// ---------------------------------------------------------------------------
// TDM: issue a 2D tensor_load_to_lds.
//   tile_d0 elems per row (contiguous), tile_d1 rows, row_stride elems between
//   row starts. Optional LDS padding: pad_code -> insert 1 DWORD after every
//   2^(pad_code+1) DWORDs stored (pad_code 3 => stride 17, 4 => stride 33).
// Issued once per calling wave (EXEC ignored); completion via TENSORcnt.
// ---------------------------------------------------------------------------
__device__ __forceinline__ void tdm_load_2d(unsigned int lds_off,
                                            const void* gptr,
                                            unsigned int tile_d0,
                                            unsigned int tile_d1,
                                            unsigned int row_stride,
                                            bool pad, unsigned int pad_code) {
    const unsigned long long ga = (unsigned long long)(uintptr_t)gptr;
    v4u g0;
    g0.x = 1u;                                         // count=1, no gather
    g0.y = lds_off;                                    // LDS byte address
    g0.z = (unsigned int)(ga & 0xffffffffu);           // global_addr[31:0]
    g0.w = (unsigned int)((ga >> 32) & 0x01ffffffu)    // global_addr[56:32]
         | (2u << 30);                                 // type = 2 (image)
    v8i g1;
    unsigned int dw0 = (2u << 16);                     // data_size = 4B
    if (pad) dw0 |= (1u << 20) | (pad_code << 22);     // pad_enable/interval, amount=1DW
    g1[0] = (int)dw0;
    g1[1] = (int)((tile_d0 & 0xffffu) << 16);          // tensor_dim0[15:0]
    g1[2] = (int)(((tile_d0 >> 16) & 0xffffu)          // tensor_dim0[31:16]
                | ((tile_d1 & 0xffffu) << 16));        // tensor_dim1[15:0]
    g1[3] = (int)(((tile_d1 >> 16) & 0xffffu)          // tensor_dim1[31:16]
                | ((tile_d0 & 0xffffu) << 16));        // tile_dim0
    g1[4] = (int)(tile_d1 & 0xffffu);                  // tile_dim1 (tile_dim2=0)
    g1[5] = (int)row_stride;                           // tensor_dim0_stride[31:0]
    g1[6] = 0;                                         // stride hi / dim1_stride lo
    g1[7] = 0;
    const v4i z4 = {0, 0, 0, 0};
#if __clang_major__ >= 23
    const v8i z8 = {0, 0, 0, 0, 0, 0, 0, 0};
    __builtin_amdgcn_tensor_load_to_lds(g0, g1, z4, z4, z8, 0);
#else
    __builtin_amdgcn_tensor_load_to_lds(g0, g1, z4, z4, 0);
#endif
}

// ---------------------------------------------------------------------------
// K1: per-(d) parameter precompute.
//   q[d,n]   = 1 - sigmoid(damp)*sigmoid(decay)
//   w[d,n]   = sigmoid(damp)*ema*proj*0.25
//   qC[d,n]  = q^16
//   T[d,m,k] = K[d, m-k] (m>=k) with K[d,j] = sum_n w q^j   (Toeplitz, 16x16)
//   W2[d,l',n] = w * q^(l'+1)                               (state projection)
// ---------------------------------------------------------------------------
__global__ void ema_params_kernel(const float* __restrict__ damp,
                                  const float* __restrict__ decay,
                                  const float* __restrict__ ema,
                                  const float* __restrict__ proj,
                                  float* __restrict__ Wq,
                                  float* __restrict__ WqC,
                                  float* __restrict__ WT,
                                  float* __restrict__ WW2) {
    __shared__ float lds_pw[NN * CCH];  // w * q^j
    __shared__ float lds_k[CCH];        // K[d, j]
    const int d = blockIdx.x;
    const int t = threadIdx.x;  // 32 threads
    if (t < NN) {
        const int n = t;
        const float p = sigmoidf_(damp[d * NN + n]);
        const float q = 1.0f - p * sigmoidf_(decay[d * NN + n]);
        const float w = p * ema[d * NN + n] * proj[d * NN + n] * 0.25f;  // 1/sqrt(16)
        Wq[d * NN + n] = q;
        float pw = w;
        for (int j = 0; j < CCH; ++j) { lds_pw[n * CCH + j] = pw; pw *= q; }
        const float q2 = q * q, q4 = q2 * q2, q8 = q4 * q4;
        WqC[d * NN + n] = q8 * q8;  // q^16
        float v = w * q;
        for (int lp = 0; lp < CCH; ++lp) { WW2[d * 256 + lp * NN + n] = v; v *= q; }
    }
    __syncthreads();
    if (t < CCH) {
        float s = 0.0f;
        for (int n = 0; n < NN; ++n) s += lds_pw[n * CCH + t];
        lds_k[t] = s;
    }
    __syncthreads();
    for (int i = 0; i < 8; ++i) {
        const int e = t + 32 * i;  // 256 entries = m*16 + k
        const int m = e >> 4, k = e & 15;
        WT[d * 256 + e] = (m >= k) ? lds_k[m - k] : 0.0f;
    }
}

// ---------------------------------------------------------------------------
// K2: per-chunk partial sums  S[c,b,d,n] = sum_j q^(15-j) * x[16c+j, b, d]
// Block = (chunk c, 32 consecutive d, all 8 b). x tile DMA'd to LDS by TDM.
// ---------------------------------------------------------------------------
__global__ void ema_chunksum_kernel(const float* __restrict__ x,
                                    const float* __restrict__ Wq,
                                    float* __restrict__ S) {
    __shared__ float lds_t[CCH * BB * 32];  // [j*8+b][dd] = 4096 floats
    __shared__ float lds_q[32 * 17];        // [dd][n], padded
    const int c = blockIdx.x;
    const int d0 = blockIdx.y * 32;
    const int t = threadIdx.x;  // 256
    const int dd = t & 31;
    if ((t >> 5) == 0) {
        // 128 rows (j*8+b) x 32 floats, row stride D, no LDS padding
        tdm_load_2d((unsigned int)(uintptr_t)lds_t,
                    x + ((size_t)c * CCH * BB) * DD + d0,
                    32u, (unsigned int)(CCH * BB), (unsigned int)DD,
                    false, 0u);
        __builtin_amdgcn_s_wait_tensorcnt(0);
    }
    for (int i = 0; i < 2; ++i) {
        const int idx = t + 256 * i;  // 0..511 = dd2*16 + n
        const int dd2 = idx >> 4, n = idx & 15;
        lds_q[dd2 * 17 + n] = Wq[(d0 + dd2) * NN + n];
    }
    __syncthreads();
    const int b = t >> 5;  // wave id -> batch
    float xv[CCH];
    #pragma unroll
    for (int j = 0; j < CCH; ++j) xv[j] = lds_t[(j * BB + b) * 32 + dd];
    float* Sp = S + (((size_t)c * BB + b) * DD + (d0 + dd)) * NN;
    for (int n = 0; n < NN; ++n) {
        const float q = lds_q[dd * 17 + n];
        float s = 0.0f;
        #pragma unroll
        for (int j = 0; j < CCH; ++j) s = fmaf(s, q, xv[j]);  // Horner
        Sp[n] = s;
    }
}

// ---------------------------------------------------------------------------
// K3: sequential scan over 256 chunks; 131072 threads (b,d,n).
// Replaces S[c] with the state ENTERING chunk c:  H_in(c) = H(c-1).
// ---------------------------------------------------------------------------
__global__ void ema_scan_kernel(const float* __restrict__ WqC,
                                float* __restrict__ S) {
    const int g = blockIdx.x * blockDim.x + threadIdx.x;  // 131072
    const int n = g & 15;
    const int d = (g >> 4) & (DD - 1);
    const int b = g >> 14;
    const float qC = WqC[d * NN + n];
    float H = 0.0f;
    const size_t stride = (size_t)BB * DD * NN;
    float* p = S + (((size_t)b * DD + d) * NN + n);
    for (int c = 0; c < NC; ++c) {
        const float sv = *p;
        *p = H;                 // store incoming state for chunk c
        H = fmaf(qC, H, sv);    // advance by one chunk
        p += stride;
    }
}

// ---------------------------------------------------------------------------
// K4: main WMMA kernel. Block = (32 consecutive d) x (chunk pair cp).
// T/W2/x tiles DMA'd to LDS by three TDM loads (with hardware padding for
// bank-conflict-free strides 17/33). Each wave computes 4 tiles (one d each):
// 16 rows l' x 16 cols (2 chunks x 8 b):
//   out_tile = T(16x16) @ X(16x16)  +  W2(16x16) @ H(16x16)
// via 8 chained v_wmma_f32_16x16x4_f32. Epilogue: + x*rw, ReLU, LDS-staged
// transpose so final stores are coalesced 128B (32 consecutive d).
// ---------------------------------------------------------------------------
__global__ void ema_out_kernel(const float* __restrict__ x,
                               const float* __restrict__ rw,
                               const float* __restrict__ WT,
                               const float* __restrict__ WW2,
                               const float* __restrict__ S,
                               float* __restrict__ out) {
    extern __shared__ float smem[];
    float* lds_T  = smem;             // 32 * 272  (16x16 padded to [m][17])
    float* lds_W2 = lds_T + 32 * 272; // 32 * 272
    float* lds_X  = lds_W2 + 32 * 272;// 256 * 33  ([ (cc*16+j)*8+b ][ dl ])
    float* lds_O  = lds_X + 256 * 33; // 256 * 33
    float* lds_H  = lds_O + 256 * 33; // 8 * 272   (per-wave [col][17])

    const int d0 = blockIdx.x * 32;
    const int cp = blockIdx.y;   // chunk pair
    const int c0 = cp * 2;
    const int t = threadIdx.x;   // 256 = 8 waves
    const int lane = t & 31, w = t >> 5;

    // DMA the three big tiles into LDS via the Tensor Data Mover.
    // Waves 0..2 each own one descriptor; padding gives strides 17/33.
    if (w == 0) {
        // 512 rows (dl*16+m) x 16 floats -> LDS stride 17 (pad every 16 DW)
        tdm_load_2d((unsigned int)(uintptr_t)lds_T, WT + (size_t)d0 * 256,
                    16u, 512u, 16u, true, 3u);
        __builtin_amdgcn_s_wait_tensorcnt(0);
    } else if (w == 1) {
        tdm_load_2d((unsigned int)(uintptr_t)lds_W2, WW2 + (size_t)d0 * 256,
                    16u, 512u, 16u, true, 3u);
        __builtin_amdgcn_s_wait_tensorcnt(0);
    } else if (w == 2) {
        // 256 rows (l*8+b) x 32 floats, row stride D -> LDS stride 33
        tdm_load_2d((unsigned int)(uintptr_t)lds_X,
                    x + ((size_t)cp * 256) * DD + d0,
                    32u, 256u, (unsigned int)DD, true, 4u);
        __builtin_amdgcn_s_wait_tensorcnt(0);
    }
    __syncthreads();

    const int col = lane & 15;   // output column (cc*8 + b)
    const int hi  = lane >> 4;   // lane half
    const int m   = lane & 15;   // A-matrix row

    for (int tt = 0; tt < 4; ++tt) {
        const int dl = w * 4 + tt;
        const int d = d0 + dl;

        // Load this tile's H_in (16 n x 16 cols) into per-wave LDS [col][n].
        {
            const int colh = lane >> 1;
            const int nb = (lane & 1) * 8;
            const float* hp =
                S + (((size_t)(c0 + (colh >> 3)) * BB + (colh & 7)) * DD + d) * NN + nb;
            const float4 h0 = *(const float4*)hp;
            const float4 h1 = *(const float4*)(hp + 4);
            float* q = &lds_H[w * 272 + colh * 17 + nb];
            q[0] = h0.x; q[1] = h0.y; q[2] = h0.z; q[3] = h0.w;
            q[4] = h1.x; q[5] = h1.y; q[6] = h1.z; q[7] = h1.w;
        }

        v8f acc = {0.f, 0.f, 0.f, 0.f, 0.f, 0.f, 0.f, 0.f};
        // Intra-chunk causal conv: acc += T @ X   (K=16 as 4 slices of 4)
        #pragma unroll
        for (int s = 0; s < 4; ++s) {
            const int k0 = 4 * s + 2 * hi;
            v2f a, b;
            a.x = lds_T[dl * 272 + m * 17 + k0];
            a.y = lds_T[dl * 272 + m * 17 + k0 + 1];
            b.x = lds_X[(((col >> 3) * 16 + k0    ) * 8 + (col & 7)) * 33 + dl];
            b.y = lds_X[(((col >> 3) * 16 + k0 + 1) * 8 + (col & 7)) * 33 + dl];
            acc = __builtin_amdgcn_wmma_f32_16x16x4_f32(false, a, false, b,
                                                        (short)0, acc, false, false);
        }
        // Boundary-state projection: acc += W2 @ H_in
        #pragma unroll
        for (int s = 0; s < 4; ++s) {
            const int k0 = 4 * s + 2 * hi;
            v2f a, b;
            a.x = lds_W2[dl * 272 + m * 17 + k0];
            a.y = lds_W2[dl * 272 + m * 17 + k0 + 1];
            b.x = lds_H[w * 272 + col * 17 + k0];
            b.y = lds_H[w * 272 + col * 17 + k0 + 1];
            acc = __builtin_amdgcn_wmma_f32_16x16x4_f32(false, a, false, b,
                                                        (short)0, acc, false, false);
        }

        // Epilogue: residual + ReLU, stage into LDS for coalesced store.
        const float rwd = rw[d];
        #pragma unroll
        for (int v = 0; v < 8; ++v) {
            const int mm = v + 8 * hi;  // C/D layout: row = vgpr + 8*(lane>=16)
            const int rowid = ((col >> 3) * 16 + mm) * 8 + (col & 7);
            const float xval = lds_X[rowid * 33 + dl];
            const float val = acc[v] + xval * rwd;
            lds_O[rowid * 33 + dl] = fmaxf(val, 0.0f);
        }
    }
    __syncthreads();

    // Coalesced writeback: per wave, 32 consecutive d at fixed (l,b).
    for (int i = 0; i < 32; ++i) {
        const int e = t + 256 * i;
        const int rowid = e >> 5, dl = e & 31;
        out[((size_t)cp * 256 + rowid) * DD + d0 + dl] = lds_O[rowid * 33 + dl];
    }
}

// ---------------------------------------------------------------------------
extern "C" void kernel_launch(void* const* d_in, const int* in_sizes, int n_in,
                              void* d_out, int out_size, void* d_ws, size_t ws_size,
                              hipStream_t stream) {
    const float* x     = (const float*)d_in[0];  // (L,B,D)
    const float* damp  = (const float*)d_in[1];  // (D,N,1)
    const float* decay = (const float*)d_in[2];  // (D,N,1)
    const float* ema   = (const float*)d_in[3];  // (D,N,1)
    const float* proj  = (const float*)d_in[4];  // (D,N)
    const float* rw    = (const float*)d_in[5];  // (D,)
    float* out = (float*)d_out;

    // Workspace layout (floats): params ~2.3MB + S 128MB
    float* wsf = (float*)d_ws;
    float* Wq  = wsf;                 // D*N
    float* WqC = Wq  + DD * NN;       // D*N
    float* WT  = WqC + DD * NN;       // D*256
    float* WW2 = WT  + DD * 256;      // D*256
    float* S   = WW2 + DD * 256;      // NC*B*D*N = 33.5M floats

    ema_params_kernel<<<DD, 32, 0, stream>>>(damp, decay, ema, proj, Wq, WqC, WT, WW2);

    dim3 g2(NC, DD / 32);
    ema_chunksum_kernel<<<g2, 256, 0, stream>>>(x, Wq, S);

    ema_scan_kernel<<<(BB * DD * NN) / 256, 256, 0, stream>>>(WqC, S);

    dim3 g4(DD / 32, NC / 2);
    const size_t shmem = (size_t)(32 * 272 * 2 + 256 * 33 * 2 + 8 * 272) * sizeof(float);
    ema_out_kernel<<<g4, 256, shmem, stream>>>(x, rw, WT, WW2, S, out);
}